// DynamicCapacityRouter_50534585205487
// MI455X (gfx1250) — compile-verified
//
#include <hip/hip_runtime.h>
#include <hip/hip_bf16.h>
#include <math.h>

#define NTOK 16384
#define NHID 2048
#define NEXP 64
#define NH4  512

typedef __attribute__((ext_vector_type(2))) float v2f;
typedef __attribute__((ext_vector_type(8))) float v8f;

// ---- workspace layout (element offsets, 4B units) ----
#define OFF_MEAN   0                       // 2048 f32 column-sum accumulator
#define OFF_ENT    2048                    // 1 f32 entropy accumulator
#define OFF_CAP    2052                    // 64 i32 capacities
#define OFF_PREFIX 2116                    // 64 u32 radix prefixes / thresholds
#define OFF_REMAIN 2180                    // 64 i32 remaining admits at threshold
#define OFF_TICKET 2244                    // 64 i32 tie-break tickets
#define OFF_HIST   2308                    // 64*2048 u32 histograms
#define OFF_PROBS  (OFF_HIST + NEXP * 2048)// 16384*64 f32 router probs
#define N_ZERO     OFF_PROBS

// ----------------------------------------------------------------------------
__global__ void zero_kernel(float* ws, int n) {
    int i = blockIdx.x * blockDim.x + threadIdx.x;
    if (i < n) ws[i] = 0.0f;
}

// ---- column sums of hidden_states for mean token -----------------------------
__global__ __launch_bounds__(256)
void colsum_kernel(const float* __restrict__ hidden, float* __restrict__ mean_acc) {
    int c  = blockIdx.x * 256 + threadIdx.x;   // column, grid.x = NHID/256
    int r0 = blockIdx.y * 256;                 // row chunk, grid.y = NTOK/256
    float s = 0.0f;
    for (int r = 0; r < 256; ++r) s += hidden[(size_t)(r0 + r) * NHID + c];
    atomicAdd(&mean_acc[c], s);
}

// ---- capacity MLP: relu(mean@W1^T+b1)@W2^T+b2 -> softmax -> clip -> floor ----
__global__ __launch_bounds__(512)
void cap_mlp_kernel(const float* __restrict__ mean_acc,
                    const float* __restrict__ w1, const float* __restrict__ b1,
                    const float* __restrict__ w2, const float* __restrict__ b2,
                    int* __restrict__ capacity) {
    __shared__ float s_mean[NHID];
    __shared__ float s_h[NH4];
    __shared__ float s_logit[NEXP];
    int tid = threadIdx.x;
    for (int i = tid; i < NHID; i += 512) s_mean[i] = mean_acc[i] * (1.0f / (float)NTOK);
    __syncthreads();
    float acc = 0.0f;
    const float* wr = w1 + (size_t)tid * NHID;
    for (int k = 0; k < NHID; ++k) acc += wr[k] * s_mean[k];
    s_h[tid] = fmaxf(acc + b1[tid], 0.0f);
    __syncthreads();
    if (tid < NEXP) {
        float a2 = 0.0f;
        const float* w2r = w2 + (size_t)tid * NH4;
        for (int k = 0; k < NH4; ++k) a2 += w2r[k] * s_h[k];
        s_logit[tid] = a2 + b2[tid];
    }
    __syncthreads();
    if (tid == 0) {
        float mx = -1e30f;
        for (int e = 0; e < NEXP; ++e) mx = fmaxf(mx, s_logit[e]);
        float s = 0.0f;
        for (int e = 0; e < NEXP; ++e) { s_logit[e] = __expf(s_logit[e] - mx); s += s_logit[e]; }
        for (int e = 0; e < NEXP; ++e) {
            float w  = s_logit[e] / s;
            float cf = 1.25f + (w - 0.5f) * (2.0f - 1.0f);
            cf = fminf(fmaxf(cf, 1.0f), 2.0f);
            capacity[e] = (int)floorf((float)NTOK * cf / (float)NEXP);
        }
    }
}

// ---- router GEMM (f32 WMMA) + softmax + entropy ------------------------------
// block = 256 threads = 8 waves; each wave computes a 16-token x 64-expert tile.
__global__ __launch_bounds__(256)
void router_gemm_kernel(const float* __restrict__ hidden,
                        const float* __restrict__ rw,
                        float* __restrict__ out_logits,
                        float* __restrict__ probs,
                        float* __restrict__ ent_accum) {
    __shared__ float s_tile[8 * 16 * NEXP];            // 32 KB
    const int wave = threadIdx.x >> 5;
    const int lane = threadIdx.x & 31;
    const int half = lane >> 4;                        // K-half selector
    const int l16  = lane & 15;                        // M (A) / N (B,C) index
    const int m0   = (blockIdx.x * 8 + wave) * 16;

    v8f acc[4];
#pragma unroll
    for (int t = 0; t < 4; ++t)
#pragma unroll
        for (int r = 0; r < 8; ++r) acc[t][r] = 0.0f;

    const float* arow = hidden + (size_t)(m0 + l16) * NHID;

    for (int k0 = 0; k0 < NHID; k0 += 4) {
        // A 16x4 f32: lane<16 -> K={k0,k0+1}; lane>=16 -> K={k0+2,k0+3}
        v2f a;
        a[0] = arow[k0 + 2 * half + 0];
        a[1] = arow[k0 + 2 * half + 1];
#pragma unroll
        for (int t = 0; t < 4; ++t) {
            const float* pb = rw + (size_t)(t * 16 + l16) * NHID + k0 + 2 * half;
            v2f b;
            b[0] = pb[0];
            b[1] = pb[1];
            acc[t] = __builtin_amdgcn_wmma_f32_16x16x4_f32(
                false, a, false, b, (short)0, acc[t], false, false);
        }
    }

    // C layout: VGPR r, lanes 0-15: token m0+r, expert l16; lanes 16-31: token m0+8+r
    float* wbase = s_tile + wave * 16 * NEXP;
#pragma unroll
    for (int t = 0; t < 4; ++t)
#pragma unroll
        for (int r = 0; r < 8; ++r)
            wbase[(r + 8 * half) * NEXP + t * 16 + l16] = acc[t][r];
    __syncthreads();

    if (lane < 16) {
        int tok = m0 + lane;
        const float* lrow = wbase + lane * NEXP;
        float mx = -1e30f;
        for (int e = 0; e < NEXP; ++e) mx = fmaxf(mx, lrow[e]);
        float s = 0.0f;
        for (int e = 0; e < NEXP; ++e) s += __expf(lrow[e] - mx);
        float inv = 1.0f / s;
        float ent = 0.0f;
        float* orow = out_logits + (size_t)tok * NEXP;
        float* prow = probs + (size_t)tok * NEXP;
        for (int e = 0; e < NEXP; ++e) {
            float lg = lrow[e];
            float p  = __expf(lg - mx) * inv;
            orow[e] = lg;
            prow[e] = p;
            ent -= p * __logf(p + 1e-8f);
        }
        atomicAdd(ent_accum, ent);
    }
}

// ---- radix-select: per-expert histogram of prob bit patterns -----------------
__global__ __launch_bounds__(256)
void hist_kernel(const float* __restrict__ probs, const unsigned* __restrict__ prefix,
                 unsigned* __restrict__ hist, int shift, int width, int use_prefix) {
    int idx = blockIdx.x * 256 + threadIdx.x;  // = t*64 + e (coalesced over probs)
    int e = idx & 63;
    unsigned bits = __float_as_uint(probs[idx]);
    if (!use_prefix || (bits >> (shift + width)) == prefix[e]) {
        unsigned bin = (bits >> shift) & ((1u << width) - 1u);
        atomicAdd(&hist[(e << 11) + bin], 1u);
    }
}

// ---- radix-select: scan one expert's histogram (descending), refine prefix ---
__global__ __launch_bounds__(256)
void select_kernel(const int* __restrict__ capacity, unsigned* __restrict__ prefix,
                   int* __restrict__ remain, unsigned* __restrict__ hist,
                   int* __restrict__ ticket, int width, int is_first, int is_last) {
    int e = blockIdx.x;
    unsigned* h = hist + (e << 11);
    if (threadIdx.x == 0) {
        int need = is_first ? capacity[e] : remain[e];
        unsigned acc = 0, above = 0;
        int selbin = 0;
        for (int b = (1 << width) - 1; b >= 0; --b) {
            unsigned c = h[b];
            if (acc + c >= (unsigned)need) { selbin = b; above = acc; break; }
            acc += c;
        }
        prefix[e] = is_first ? (unsigned)selbin : ((prefix[e] << width) | (unsigned)selbin);
        remain[e] = need - (int)above;
        if (is_last) ticket[e] = 0;
    }
    __syncthreads();
    for (int b = threadIdx.x; b < 2048; b += 256) h[b] = 0;  // re-zero for next level
}

// ---- final pass: in_topk test, max-expert wins, weights, entropy write -------
__global__ __launch_bounds__(256)
void finalize_kernel(const float* __restrict__ probs, const unsigned* __restrict__ prefix,
                     const int* __restrict__ remain, int* __restrict__ ticket,
                     const float* __restrict__ ent_acc, float* __restrict__ d_out) {
    __shared__ unsigned s_th[NEXP];
    __shared__ int s_rm[NEXP];
    if (threadIdx.x < NEXP) {
        s_th[threadIdx.x] = prefix[threadIdx.x];
        s_rm[threadIdx.x] = remain[threadIdx.x];
    }
    __syncthreads();
    int t = blockIdx.x * 256 + threadIdx.x;
    const float* prow = probs + (size_t)t * NEXP;
    int sel = -1;
    float wv = 0.0f;
    for (int e = 0; e < NEXP; ++e) {
        float p = prow[e];
        unsigned b = __float_as_uint(p);
        bool in = b > s_th[e];
        if (!in && b == s_th[e]) in = (atomicAdd(&ticket[e], 1) < s_rm[e]);
        if (in) { sel = e; wv = p; }
    }
    d_out[(size_t)NTOK * NEXP + t]        = (float)(sel >= 0 ? sel : 0);
    d_out[(size_t)NTOK * NEXP + NTOK + t] = (sel >= 0 ? wv : 0.0f);
    if (t == 0) d_out[(size_t)NTOK * NEXP + 2 * NTOK] = ent_acc[0] * (1.0f / (float)NTOK);
}

// ----------------------------------------------------------------------------
extern "C" void kernel_launch(void* const* d_in, const int* in_sizes, int n_in,
                              void* d_out, int out_size, void* d_ws, size_t ws_size,
                              hipStream_t stream) {
    const float* hidden = (const float*)d_in[0];
    const float* rw     = (const float*)d_in[1];
    const float* w1     = (const float*)d_in[2];
    const float* b1     = (const float*)d_in[3];
    const float* w2     = (const float*)d_in[4];
    const float* b2     = (const float*)d_in[5];
    float* out = (float*)d_out;

    float*    ws_f    = (float*)d_ws;
    float*    mean_a  = ws_f + OFF_MEAN;
    float*    ent_a   = ws_f + OFF_ENT;
    int*      cap     = (int*)(ws_f + OFF_CAP);
    unsigned* prefix  = (unsigned*)(ws_f + OFF_PREFIX);
    int*      remain  = (int*)(ws_f + OFF_REMAIN);
    int*      ticket  = (int*)(ws_f + OFF_TICKET);
    unsigned* hist    = (unsigned*)(ws_f + OFF_HIST);
    float*    probs   = ws_f + OFF_PROBS;

    // 0) zero accumulators / histograms / tickets
    zero_kernel<<<(N_ZERO + 255) / 256, 256, 0, stream>>>(ws_f, N_ZERO);
    // 1) column sums for mean token
    colsum_kernel<<<dim3(NHID / 256, NTOK / 256), 256, 0, stream>>>(hidden, mean_a);
    // 2) capacity MLP
    cap_mlp_kernel<<<1, 512, 0, stream>>>(mean_a, w1, b1, w2, b2, cap);
    // 3) router GEMM (WMMA f32) + softmax + entropy + probs
    router_gemm_kernel<<<NTOK / 128, 256, 0, stream>>>(hidden, rw, out, probs, ent_a);
    // 4) 3-level radix select per expert (11 + 11 + 10 bits = exact f32 threshold)
    const int nhist = (NTOK * NEXP) / 256;
    hist_kernel<<<nhist, 256, 0, stream>>>(probs, prefix, hist, 21, 11, 0);
    select_kernel<<<NEXP, 256, 0, stream>>>(cap, prefix, remain, hist, ticket, 11, 1, 0);
    hist_kernel<<<nhist, 256, 0, stream>>>(probs, prefix, hist, 10, 11, 1);
    select_kernel<<<NEXP, 256, 0, stream>>>(cap, prefix, remain, hist, ticket, 11, 0, 0);
    hist_kernel<<<nhist, 256, 0, stream>>>(probs, prefix, hist, 0, 10, 1);
    select_kernel<<<NEXP, 256, 0, stream>>>(cap, prefix, remain, hist, ticket, 10, 0, 1);
    // 5) selections / weights / entropy out
    finalize_kernel<<<NTOK / 256, 256, 0, stream>>>(probs, prefix, remain, ticket, ent_a, out);
}